// BaseObjectDetector_48421461295793
// MI455X (gfx1250) — compile-verified
//
#include <hip/hip_runtime.h>
#include <hip/hip_bf16.h>
#include <float.h>

// ---------------- problem constants (match reference) ----------------
#define BATCH    16
#define NA       25200          // anchors per image
#define NCH      85             // 4 box + 1 obj + 80 cls
#define NC       80
#define TOPK     1024
#define MAXDET   300
#define CONF_TH  0.6f
#define IOU_TH   0.45f
#define MAX_WH   4096.0f
#define BINS     4096
#define CAND     2048           // candidate cap before exact sort

typedef __attribute__((ext_vector_type(2))) float v2f;
typedef __attribute__((ext_vector_type(8))) float v8f;

// =====================================================================
// Kernel A: decode. One streaming pass over prediction (137 MB -> BW bound).
// Writes per-anchor score (conf or -1), class id, xyxy box.
// =====================================================================
__global__ void nms_decode(const float* __restrict__ pred,
                           float* __restrict__ score,
                           int*   __restrict__ cls,
                           float* __restrict__ box)
{
    int g = blockIdx.x * blockDim.x + threadIdx.x;
    if (g >= BATCH * NA) return;
    const float* p = pred + (size_t)g * NCH;

    float cx = p[0], cy = p[1], w = p[2], h = p[3];
    float obj = p[4];

    float best = -FLT_MAX;
    int   bi   = 0;
#pragma unroll 8
    for (int c = 0; c < NC; ++c) {
        float v = obj * p[5 + c];
        if (v > best) { best = v; bi = c; }   // strict > keeps first index (argmax)
    }
    bool valid = (obj > CONF_TH) && (best > CONF_TH);

    float* b = box + (size_t)g * 4;
    b[0] = cx - 0.5f * w;
    b[1] = cy - 0.5f * h;
    b[2] = cx + 0.5f * w;
    b[3] = cy + 0.5f * h;
    cls[g]   = bi;
    score[g] = valid ? best : -1.0f;
}

// =====================================================================
// Kernel B: per-image top-1024 (sorted desc, ties by low index, matching
// lax.top_k). Histogram over (CONF_TH,1.0] -> threshold bin -> compact
// candidates -> exact LDS bitonic sort of 2048 -> emit top 1024.
// One block (256 threads) per image.
// =====================================================================
__device__ __forceinline__ int score_bin(float s)
{
    if (!(s > CONF_TH)) return 0;                       // invalid / -1 fill
    int b = (int)((s - CONF_TH) * ((float)BINS / (1.0f - CONF_TH)));
    if (b < 1) b = 1;
    if (b > BINS - 1) b = BINS - 1;
    return b;
}

__global__ void nms_select(const float* __restrict__ score,
                           const int*   __restrict__ cls,
                           const float* __restrict__ box,
                           float* __restrict__ selScore,
                           int*   __restrict__ selIdx,
                           float* __restrict__ selBox,
                           float* __restrict__ selNms,
                           int*   __restrict__ selCls)
{
    __shared__ int   hist[BINS];
    __shared__ float ss[CAND];
    __shared__ int   si[CAND];
    __shared__ int   s_tb, s_cnt;

    const int img = blockIdx.x;
    const int tid = threadIdx.x;
    const int T   = blockDim.x;
    const float* sc = score + (size_t)img * NA;

    for (int i = tid; i < BINS; i += T) hist[i] = 0;
    __syncthreads();

    for (int a = tid; a < NA; a += T)
        atomicAdd(&hist[score_bin(sc[a])], 1);
    __syncthreads();

    if (tid == 0) {
        int cum = 0, tb = 0;
        for (int bn = BINS - 1; bn >= 0; --bn) {
            cum += hist[bn];
            if (cum >= TOPK) { tb = bn; break; }
        }
        s_tb = tb;
        s_cnt = 0;
    }
    __syncthreads();

    const int tb = s_tb;
    for (int a = tid; a < NA; a += T) {
        float s = sc[a];
        if (score_bin(s) >= tb) {
            int pos = atomicAdd(&s_cnt, 1);
            if (pos < CAND) { ss[pos] = s; si[pos] = a; }
        }
    }
    __syncthreads();

    int cnt = s_cnt; if (cnt > CAND) cnt = CAND;
    for (int i = cnt + tid; i < CAND; i += T) { ss[i] = -3.0e38f; si[i] = -1; }
    __syncthreads();

    // bitonic sort: "precedes" = (higher score) then (lower anchor index)
    for (unsigned k = 2; k <= CAND; k <<= 1) {
        for (unsigned j = k >> 1; j > 0; j >>= 1) {
            for (unsigned i = tid; i < CAND; i += T) {
                unsigned l = i ^ j;
                if (l > i) {
                    bool dir = ((i & k) == 0);
                    float a0 = ss[i], a1 = ss[l];
                    int   i0 = si[i], i1 = si[l];
                    bool lPrec = (a1 > a0) || (a1 == a0 && ((unsigned)i1 < (unsigned)i0));
                    if (lPrec == dir) {
                        ss[i] = a1; ss[l] = a0;
                        si[i] = i1; si[l] = i0;
                    }
                }
            }
            __syncthreads();
        }
    }

    for (int r = tid; r < TOPK; r += T) {
        float s = ss[r];
        int   a = si[r];
        int o = img * TOPK + r;
        if (a >= 0 && a < NA) {
            const float* b = box + ((size_t)img * NA + a) * 4;
            int c = cls[(size_t)img * NA + a];
            float off = (float)c * MAX_WH;     // AGNOSTIC == false
            selScore[o] = s;
            selIdx[o]   = a;
            selCls[o]   = c;
            float* sb = selBox + (size_t)o * 4;
            float* sn = selNms + (size_t)o * 4;
            sb[0] = b[0]; sb[1] = b[1]; sb[2] = b[2]; sb[3] = b[3];
            sn[0] = b[0] + off; sn[1] = b[1] + off;
            sn[2] = b[2] + off; sn[3] = b[3] + off;
        } else {
            selScore[o] = -1.0f; selIdx[o] = 0; selCls[o] = 0;
            float* sb = selBox + (size_t)o * 4;
            float* sn = selNms + (size_t)o * 4;
            sb[0]=sb[1]=sb[2]=sb[3]=0.f;
            sn[0]=sn[1]=sn[2]=sn[3]=0.f;
        }
    }
}

// =====================================================================
// Kernel C: build the 1024x1024 suppression bitmask per image.
// Each wave owns a 16-row x 32-col region (exclusive mask words).
// The pairwise term (area_i + area_j) of the IoU denominator is a rank-2
// outer product computed with V_WMMA_F32_16X16X4_F32:
//   A(16x4) = [area_i, 1, 0, 0],  B(4x16) = [1...; area_j...; 0; 0]
// Intersection + compare done in VALU; bits packed with wave32 ballots
// that map 1:1 onto the WMMA C-matrix row layout.
// =====================================================================
__global__ void nms_mask(const float* __restrict__ selNms,
                         unsigned* __restrict__ mask)
{
    const int wave = (blockIdx.x * blockDim.x + threadIdx.x) >> 5;
    const int lane = threadIdx.x & 31;
    const int img     = wave >> 11;        // 2048 waves / image
    const int rem     = wave & 2047;
    const int rowTile = rem >> 5;          // 0..63  (16 rows each)
    const int colWord = rem & 31;          // 0..31  (32 cols each)
    const int rowBase = rowTile * 16;
    const int colBase = colWord * 32;
    const float* nb = selNms + (size_t)img * TOPK * 4;

    // ---- WMMA fragments ----
    v2f a, b0, b1;
    if (lane < 16) {
        const float* rb = nb + (size_t)(rowBase + lane) * 4;
        a.x = (rb[2] - rb[0]) * (rb[3] - rb[1]);   // K=0: area_i
        a.y = 1.0f;                                // K=1: ones
        const float* c0 = nb + (size_t)(colBase + lane) * 4;
        b0.x = 1.0f;                               // K=0 row: ones
        b0.y = (c0[2] - c0[0]) * (c0[3] - c0[1]);  // K=1 row: area_j
        const float* c1 = nb + (size_t)(colBase + 16 + lane) * 4;
        b1.x = 1.0f;
        b1.y = (c1[2] - c1[0]) * (c1[3] - c1[1]);
    } else {                                       // K=2,3 -> zero pad
        a.x = a.y = 0.0f;
        b0.x = b0.y = 0.0f;
        b1.x = b1.y = 0.0f;
    }
    v8f cz = {};
    v8f d0 = __builtin_amdgcn_wmma_f32_16x16x4_f32(false, a, false, b0,
                                                   (short)0, cz, false, false);
    v8f d1 = __builtin_amdgcn_wmma_f32_16x16x4_f32(false, a, false, b1,
                                                   (short)0, cz, false, false);

    // per-lane fixed columns
    const int j0 = colBase + (lane & 15);
    const int j1 = j0 + 16;
    const float* cb0 = nb + (size_t)j0 * 4;
    const float* cb1 = nb + (size_t)j1 * 4;
    float c0x1 = cb0[0], c0y1 = cb0[1], c0x2 = cb0[2], c0y2 = cb0[3];
    float c1x1 = cb1[0], c1y1 = cb1[1], c1x2 = cb1[2], c1y2 = cb1[3];

    const int mrow = (lane < 16) ? 0 : 8;
    unsigned ball0[8], ball1[8];
#pragma unroll
    for (int e = 0; e < 8; ++e) {
        int i = rowBase + mrow + e;
        const float* rb = nb + (size_t)i * 4;
        float rx1 = rb[0], ry1 = rb[1], rx2 = rb[2], ry2 = rb[3];

        float iw0 = fminf(rx2, c0x2) - fmaxf(rx1, c0x1);
        float ih0 = fminf(ry2, c0y2) - fmaxf(ry1, c0y1);
        float in0 = fmaxf(iw0, 0.0f) * fmaxf(ih0, 0.0f);
        float iou0 = in0 / (d0[e] - in0 + 1e-9f);        // d0[e] = area_i+area_j
        bool p0 = (iou0 > IOU_TH) && (j0 > i);
        ball0[e] = (unsigned)__ballot(p0);

        float iw1 = fminf(rx2, c1x2) - fmaxf(rx1, c1x1);
        float ih1 = fminf(ry2, c1y2) - fmaxf(ry1, c1y1);
        float in1 = fmaxf(iw1, 0.0f) * fmaxf(ih1, 0.0f);
        float iou1 = in1 / (d1[e] - in1 + 1e-9f);
        bool p1 = (iou1 > IOU_TH) && (j1 > i);
        ball1[e] = (unsigned)__ballot(p1);
    }

    if (lane == 0) {
        unsigned* mw = mask + ((size_t)img * TOPK + rowBase) * 32 + colWord;
#pragma unroll
        for (int r = 0; r < 8; ++r) {
            // ballot bits [15:0] = row r cols 0..15, [31:16] = row r+8 cols 0..15
            unsigned wlo = (ball0[r] & 0xFFFFu) | ((ball1[r] & 0xFFFFu) << 16);
            unsigned whi = ((ball0[r] >> 16) & 0xFFFFu) |
                           (((ball1[r] >> 16) & 0xFFFFu) << 16);
            mw[(size_t)r * 32]       = wlo;
            mw[(size_t)(r + 8) * 32] = whi;
        }
    }
}

// =====================================================================
// Kernel D: greedy sequential suppression. One wave per image; each lane
// holds the suppressed bits of its 32 columns in a register; the row-i
// bit is fetched with a wave32 shuffle.
// =====================================================================
__global__ void nms_greedy(const float* __restrict__ selScore,
                           const unsigned* __restrict__ mask,
                           unsigned* __restrict__ keep)
{
    const int img  = blockIdx.x;
    const int lane = threadIdx.x;
    unsigned sup = 0;
    const unsigned* m = mask + (size_t)img * TOPK * 32;
    const float* sc = selScore + (size_t)img * TOPK;

    for (int i = 0; i < TOPK; ++i) {
        unsigned ow = (unsigned)__shfl((int)sup, i >> 5, 32);
        bool supi = (ow >> (i & 31)) & 1u;
        bool keepi = (sc[i] > CONF_TH) && !supi;
        if (keepi) sup |= m[(size_t)i * 32 + lane];
        if (lane == 0) keep[(size_t)img * TOPK + i] = keepi ? 1u : 0u;
    }
}

// =====================================================================
// Kernel E: final top-300 of kept scores (bitonic, ties by position) and
// output assembly: det(16,300,6), valid(16,300), logits(16,300,80).
// =====================================================================
__global__ void nms_finalize(const float* __restrict__ selScore,
                             const int*   __restrict__ selIdx,
                             const float* __restrict__ selBox,
                             const int*   __restrict__ selCls,
                             const unsigned* __restrict__ keep,
                             const float* __restrict__ logits,
                             float* __restrict__ det,
                             float* __restrict__ validOut,
                             float* __restrict__ logitsOut)
{
    __shared__ float ss[TOPK];
    __shared__ int   sr[TOPK];
    const int img = blockIdx.x;
    const int tid = threadIdx.x;
    const int T   = blockDim.x;

    for (int r = tid; r < TOPK; r += T) {
        unsigned kp = keep[(size_t)img * TOPK + r];
        ss[r] = kp ? selScore[(size_t)img * TOPK + r] : -1.0f;
        sr[r] = r;
    }
    __syncthreads();

    for (unsigned k = 2; k <= TOPK; k <<= 1) {
        for (unsigned j = k >> 1; j > 0; j >>= 1) {
            for (unsigned i = tid; i < TOPK; i += T) {
                unsigned l = i ^ j;
                if (l > i) {
                    bool dir = ((i & k) == 0);
                    float a0 = ss[i], a1 = ss[l];
                    int   i0 = sr[i], i1 = sr[l];
                    bool lPrec = (a1 > a0) || (a1 == a0 && i1 < i0);
                    if (lPrec == dir) {
                        ss[i] = a1; ss[l] = a0;
                        sr[i] = i1; sr[l] = i0;
                    }
                }
            }
            __syncthreads();
        }
    }

    for (int r = tid; r < MAXDET; r += T) {
        int   p = sr[r];
        float s = ss[r];
        bool  v = s > CONF_TH;
        float* dp = det + ((size_t)img * MAXDET + r) * 6;
        if (v) {
            const float* b = selBox + ((size_t)img * TOPK + p) * 4;
            dp[0] = b[0]; dp[1] = b[1]; dp[2] = b[2]; dp[3] = b[3];
            dp[4] = selScore[(size_t)img * TOPK + p];
            dp[5] = (float)selCls[(size_t)img * TOPK + p];
        } else {
            dp[0]=dp[1]=dp[2]=dp[3]=dp[4]=dp[5] = 0.0f;
        }
        validOut[(size_t)img * MAXDET + r] = v ? 1.0f : 0.0f;
    }
    __syncthreads();

    for (int t = tid; t < MAXDET * NC; t += T) {
        int r = t / NC, c = t % NC;
        int   p = sr[r];
        bool  v = ss[r] > CONF_TH;
        float out = 0.0f;
        if (v) {
            int a = selIdx[(size_t)img * TOPK + p];
            out = logits[((size_t)img * NA + a) * NC + c];
        }
        logitsOut[((size_t)img * MAXDET + r) * NC + c] = out;
    }
}

// =====================================================================
// launch
// =====================================================================
extern "C" void kernel_launch(void* const* d_in, const int* in_sizes, int n_in,
                              void* d_out, int out_size, void* d_ws, size_t ws_size,
                              hipStream_t stream)
{
    const float* pred   = (const float*)d_in[0];   // (16,25200,85) f32
    const float* logits = (const float*)d_in[1];   // (16,25200,80) f32

    // ---- carve workspace (~12.6 MB, all fully rewritten every call) ----
    char* w = (char*)d_ws;
    float*    score    = (float*)w;                 w += (size_t)BATCH * NA * 4;
    int*      cls      = (int*)w;                   w += (size_t)BATCH * NA * 4;
    float*    box      = (float*)w;                 w += (size_t)BATCH * NA * 4 * 4;
    float*    selScore = (float*)w;                 w += (size_t)BATCH * TOPK * 4;
    int*      selIdx   = (int*)w;                   w += (size_t)BATCH * TOPK * 4;
    float*    selBox   = (float*)w;                 w += (size_t)BATCH * TOPK * 4 * 4;
    float*    selNms   = (float*)w;                 w += (size_t)BATCH * TOPK * 4 * 4;
    int*      selCls   = (int*)w;                   w += (size_t)BATCH * TOPK * 4;
    unsigned* mask     = (unsigned*)w;              w += (size_t)BATCH * TOPK * 32 * 4;
    unsigned* keep     = (unsigned*)w;              w += (size_t)BATCH * TOPK * 4;

    // ---- outputs: det | valid | logits, concatenated flat ----
    float* det       = (float*)d_out;
    float* validOut  = det + (size_t)BATCH * MAXDET * 6;
    float* logitsOut = validOut + (size_t)BATCH * MAXDET;

    // A: streaming decode (bandwidth-bound pass over 137 MB)
    {
        int n = BATCH * NA;
        nms_decode<<<(n + 255) / 256, 256, 0, stream>>>(pred, score, cls, box);
    }
    // B: per-image top-1024 (hist + compact + bitonic)
    nms_select<<<BATCH, 256, 0, stream>>>(score, cls, box,
                                          selScore, selIdx, selBox, selNms, selCls);
    // C: suppression bitmask via WMMA outer-sum + VALU IoU + ballots
    //    waves/image = 64 rowTiles * 32 colWords = 2048; 8 waves/block
    nms_mask<<<BATCH * 256, 256, 0, stream>>>(selNms, mask);
    // D: greedy scan, one wave per image
    nms_greedy<<<BATCH, 32, 0, stream>>>(selScore, mask, keep);
    // E: top-300 + output assembly
    nms_finalize<<<BATCH, 256, 0, stream>>>(selScore, selIdx, selBox, selCls,
                                            keep, logits, det, validOut, logitsOut);
}